// CTCLayer_33784212750819
// MI455X (gfx1250) — compile-verified
//
#include <hip/hip_runtime.h>
#include <hip/hip_bf16.h>
#include <math.h>
#include <stdint.h>

// Problem constants (from reference setup_inputs()):
#define CTC_B 32
#define CTC_T 2048
#define CTC_C 256
#define CTC_L 256
#define CTC_S (2 * CTC_L + 1)   // 513 extended states
#define NEGV (-1e30f)
#define EPSV (1e-7f)
#define NTHREADS 512            // 16 wave32 per block; thread 0 also owns s=512
#define NBUF 4                  // row pipeline depth (prefetch distance 3)

#define AS1 __attribute__((address_space(1)))
#define AS3 __attribute__((address_space(3)))

typedef int v4i __attribute__((ext_vector_type(4)));

#if defined(__gfx1250__) && __has_builtin(__builtin_amdgcn_global_load_async_to_lds_b128)
#define HAVE_ASYNC_LDS 1
#else
#define HAVE_ASYNC_LDS 0
#endif

template <int N>
__device__ __forceinline__ void wait_asynccnt() {
#if defined(__gfx1250__)
#if __has_builtin(__builtin_amdgcn_s_wait_asynccnt)
  __builtin_amdgcn_s_wait_asynccnt(N);
#else
  asm volatile("s_wait_asynccnt %0" ::"n"(N) : "memory");
#endif
#endif
}

__device__ __forceinline__ void async_row_load(const float* gsrc, float* ldsdst) {
#if HAVE_ASYNC_LDS
  // 16 bytes per lane, 64 lanes (2 waves) -> one 1 KB row; tracked by ASYNCcnt.
  __builtin_amdgcn_global_load_async_to_lds_b128(
      (AS1 v4i*)(gsrc), (AS3 v4i*)(ldsdst), /*offset=*/0, /*cpol=*/0);
#else
  const float4 v = *(const float4*)gsrc;
  *(float4*)ldsdst = v;
#endif
}

__global__ __launch_bounds__(NTHREADS, 1)
void ctc_fwd_kernel(const int* __restrict__ y_true,
                    const float* __restrict__ y_pred,
                    float* __restrict__ out) {
  const int b   = blockIdx.x;
  const int tid = threadIdx.x;
  const int blank = CTC_C - 1;

  // alpha ping-pong buffers, with 2 NEG pad slots at the front so that
  // alpha[s-1], alpha[s-2] are plain reads: buf[2+s-1], buf[2+s-2].
  __shared__ float alphaA[CTC_S + 2];
  __shared__ float alphaB[CTC_S + 2];
  __shared__ float row[NBUF][CTC_C];   // 4-deep pipelined y_pred rows (1 KB each)

  // ---- Per-thread state metadata (registers): s = tid, thread 0 also s=512.
  int  z0;
  bool al0;
  {
    const int s = tid;
    if (s & 1) {
      const int li = s >> 1;
      z0 = y_true[b * CTC_L + li];                      // label state
      const int zm2 = (li > 0) ? y_true[b * CTC_L + li - 1] : -1;
      al0 = (z0 != zm2);                                // skip legal iff labels differ
    } else {
      z0  = blank;                                      // blank state
      al0 = false;
    }
  }
  // s = 512 is even -> blank, skip never allowed (handled inline by thread 0).

  const float* gsrc = y_pred + (size_t)b * CTC_T * CTC_C;
  const bool ld = (tid < CTC_C / 4);    // waves 0,1 own the row DMA

  // ---- Setup: row 0 synchronous; rows 1..3 async in flight.
  if (ld) {
    const float4 v = *(const float4*)(gsrc + 4 * tid);
    *(float4*)&row[0][4 * tid] = v;
    async_row_load(gsrc + (size_t)1 * CTC_C + 4 * tid, &row[1][4 * tid]);
    async_row_load(gsrc + (size_t)2 * CTC_C + 4 * tid, &row[2][4 * tid]);
    async_row_load(gsrc + (size_t)3 * CTC_C + 4 * tid, &row[3][4 * tid]);
  }
  if (tid < 2) { alphaA[tid] = NEGV; alphaB[tid] = NEGV; }
  __syncthreads();

  // alpha0: only s < 2 live.
  {
    const float p = row[0][z0];
    alphaA[2 + tid] = (tid < 2) ? __logf(p + EPSV) : NEGV;
    if (tid == 0) alphaA[2 + 512] = NEGV;
  }
  if (ld) wait_asynccnt<2>();   // row 1 resident (in-order completion), 2,3 in flight
  __syncthreads();

  // ---- Main DP: one block barrier per timestep; 3-row-deep DMA prefetch
  //      hides HBM/L2 latency behind the serial LSE chain.
  float* aPrev = alphaA;
  float* aNext = alphaB;
  for (int t = 1; t < CTC_T; ++t) {
    const int cur = t & (NBUF - 1);

    // Prefetch row t+3 into the buffer that held row t-1 (consumed last step).
    if (t + 3 < CTC_T && ld)
      async_row_load(gsrc + (size_t)(t + 3) * CTC_C + 4 * tid,
                     &row[(t + 3) & (NBUF - 1)][4 * tid]);

    // State s = tid.
    {
      const float a0 = aPrev[2 + tid];
      const float a1 = aPrev[1 + tid];
      const float a2 = al0 ? aPrev[tid] : NEGV;
      const float m  = fmaxf(a0, fmaxf(a1, a2));
      const float lse = m + __logf(__expf(a0 - m) + __expf(a1 - m) + __expf(a2 - m));
      const float lp  = __logf(row[cur][z0] + EPSV);
      aNext[2 + tid] = lp + lse;
    }
    // State s = 512 (even -> blank, no skip), owned by thread 0.
    if (tid == 0) {
      const float a0 = aPrev[2 + 512];
      const float a1 = aPrev[1 + 512];
      const float m  = fmaxf(a0, a1);
      const float lse = m + __logf(__expf(a0 - m) + __expf(a1 - m));
      const float lp  = __logf(row[cur][blank] + EPSV);
      aNext[2 + 512] = lp + lse;
    }

    // Guarantee row t+1 resident before the barrier publishes it to all waves:
    // outstanding after wait = {t+2, t+3} ∩ [0,T), i.e. <=2.
    if (ld) {
      if (t + 3 < CTC_T)       wait_asynccnt<2>();
      else if (t + 2 < CTC_T)  wait_asynccnt<1>();
      else if (t + 1 < CTC_T)  wait_asynccnt<0>();
    }
    __syncthreads();

    float* tmp = aPrev; aPrev = aNext; aNext = tmp;
  }

  // ---- Final: -logsumexp(alpha[S-1], alpha[S-2]).
  if (tid == 0) {
    const float last  = aPrev[2 + CTC_S - 1];
    const float last2 = aPrev[2 + CTC_S - 2];
    const float m  = fmaxf(last, last2);
    const float ll = m + __logf(__expf(last - m) + __expf(last2 - m));
    out[b] = -ll;
  }
}

extern "C" void kernel_launch(void* const* d_in, const int* in_sizes, int n_in,
                              void* d_out, int out_size, void* d_ws, size_t ws_size,
                              hipStream_t stream) {
  (void)in_sizes; (void)n_in; (void)out_size; (void)d_ws; (void)ws_size;
  const int*   y_true = (const int*)d_in[0];    // [B, L] int32
  const float* y_pred = (const float*)d_in[1];  // [B, T, C] float32
  float*       out    = (float*)d_out;          // [B, 1] float32

  ctc_fwd_kernel<<<dim3(CTC_B), dim3(NTHREADS), 0, stream>>>(y_true, y_pred, out);
}